// SPDNet_27805618274875
// MI455X (gfx1250) — compile-verified
//
#include <hip/hip_runtime.h>
#include <hip/hip_bf16.h>

typedef __attribute__((ext_vector_type(2))) float v2f;
typedef __attribute__((ext_vector_type(8))) float v8f;

#define EPS 1e-4f
#define NS_ITERS 12
#define WAVES_PER_WG 4
#define NUM_WGS 2048   // persistent: 8192 waves, ~4 batch elements each

// Padded leading dims: all 32 lanes hit distinct LDS banks, rows stay 8B-aligned.
#define LD32 34   // 32x32 tiles
#define LDT  68   // T: 32x64
#define LDW  40   // W: 64x32

// ---- WMMA f32 16x16x4 fragment helpers (layouts per CDNA5 ISA 7.12.2) ----
// Row-pattern fragment: lane&15 = row, K = kb + 2*(lane>>4) + vgpr.
// Serves as the A operand, and ALSO as the B operand of any SYMMETRIC matrix
// (B[k][n] = M[n][k] = consecutive floats of row n) -> single ds_load_b64.
__device__ __forceinline__ v2f frag_row2(const float* m_, int ld, int r0, int kb, int lane) {
  int r = r0 + (lane & 15);
  int k = kb + ((lane >> 4) << 1);
  return *(const v2f*)(m_ + r * ld + k);
}
// B fragment for a general row-major [K x N] matrix (two b32, lane-coalesced).
__device__ __forceinline__ v2f frag_col(const float* m_, int ld, int kb, int n0, int lane) {
  int n = n0 + (lane & 15);
  int k = kb + ((lane >> 4) << 1);
  v2f b; b.x = m_[k * ld + n]; b.y = m_[(k + 1) * ld + n];
  return b;
}
__device__ __forceinline__ v8f wmma4(v2f a, v2f b, v8f c) {
  return __builtin_amdgcn_wmma_f32_16x16x4_f32(false, a, false, b, (short)0, c, false, false);
}
// C/D 16x16 f32: VGPR r -> row m0 + 8*(lane>>4) + r, col n0 + (lane&15)
__device__ __forceinline__ void store_c(float* m_, int ld, int m0, int n0, int lane, v8f c) {
  int n = n0 + (lane & 15);
  int mb = m0 + ((lane >> 4) << 3);
#pragma unroll
  for (int r = 0; r < 8; ++r) m_[(mb + r) * ld + n] = c[r];
}
// Transposed store: mirror an off-diagonal tile of a symmetric result.
__device__ __forceinline__ void store_c_t(float* m_, int ld, int m0, int n0, int lane, v8f c) {
  int n = n0 + (lane & 15);
  int mb = m0 + ((lane >> 4) << 3);
#pragma unroll
  for (int r = 0; r < 8; ++r) m_[n * ld + (mb + r)] = c[r];
}
// Fused NS writeback: Q = 1.5*P - 0.5*acc (+ mirror for off-diagonal tile)
__device__ __forceinline__ void store_ns(float* Qm, const float* Pm, int m0, int n0,
                                         int lane, v8f c, bool mirror) {
  int n = n0 + (lane & 15);
  int mb = m0 + ((lane >> 4) << 3);
#pragma unroll
  for (int r = 0; r < 8; ++r) {
    int row = mb + r;
    float v = 1.5f * Pm[row * LD32 + n] - 0.5f * c[r];
    Qm[row * LD32 + n] = v;
    if (mirror) Qm[n * LD32 + row] = v;
  }
}
// Final writeback: R = 0.5*(Y + EPS*I + acc) (+ mirror)
__device__ __forceinline__ void store_fin(float* Rm, const float* Ym, int m0, int n0,
                                          int lane, v8f c, bool mirror) {
  int n = n0 + (lane & 15);
  int mb = m0 + ((lane >> 4) << 3);
#pragma unroll
  for (int r = 0; r < 8; ++r) {
    int row = mb + r;
    float v = 0.5f * (Ym[row * LD32 + n] + (row == n ? EPS : 0.f) + c[r]);
    Rm[row * LD32 + n] = v;
    if (mirror) Rm[n * LD32 + row] = v;
  }
}

__global__ __launch_bounds__(WAVES_PER_WG * 32) void spdnet_wmma_kernel(
    const float* __restrict__ x, const float* __restrict__ Wg,
    const float* __restrict__ cls_w, const float* __restrict__ cls_b,
    float* __restrict__ out, int nB) {
  extern __shared__ float smem[];
  const int lane = threadIdx.x & 31;
  const int wave = threadIdx.x >> 5;

  // One shared, bank-padded copy of W (64x32 -> ld 40) per workgroup
  float* Wlds = smem;  // 64*LDW
  for (int i = threadIdx.x; i < 64 * 32; i += blockDim.x)
    Wlds[(i >> 5) * LDW + (i & 31)] = Wg[i];
  __syncthreads();

  const int WSLICE = 32 * LDT + 4 * 32 * LD32;
  float* base = smem + 64 * LDW + wave * WSLICE;
  float* T  = base;                 // 32 x LDT
  float* Y  = base + 32 * LDT;      // 32 x LD32
  float* Pb = Y + 32 * LD32;
  float* Qb = Pb + 32 * LD32;
  float* ZZ = Qb + 32 * LD32;

  const int wid     = blockIdx.x * WAVES_PER_WG + wave;
  const int nWaves  = NUM_WGS * WAVES_PER_WG;

  // Warm the cache for this wave's first tile.
  if (wid < nB) {
    const float* x0p = x + (size_t)wid * 4096;
    for (int off = lane * 16; off < 4096; off += 32 * 16)
      __builtin_prefetch(x0p + off, 0, 1);
  }

  // Persistent-wave loop: all branches wave-uniform -> EXEC all-1s at every WMMA.
  for (int b = wid; b < nB; b += nWaves) {
    const float* xb = x + (size_t)b * 4096;
    float* P = Pb;
    float* Q = Qb;

    // ---- Step 1: T = W^T * x (32x64). 2x2 register blocking; x streamed once. ----
    for (int ng = 0; ng < 64; ng += 32) {
      v8f a00 = {}, a01 = {}, a10 = {}, a11 = {};
      for (int kb = 0; kb < 64; kb += 4) {
        v2f wa0 = frag_col(Wlds, LDW, kb, 0, lane);   // (W^T)[m][k] = W[k][m]
        v2f wa1 = frag_col(Wlds, LDW, kb, 16, lane);
        v2f x0  = frag_col(xb, 64, kb, ng, lane);      // coalesced global
        v2f x1  = frag_col(xb, 64, kb, ng + 16, lane);
        a00 = wmma4(wa0, x0, a00);
        a01 = wmma4(wa0, x1, a01);
        a10 = wmma4(wa1, x0, a10);
        a11 = wmma4(wa1, x1, a11);
      }
      store_c(T, LDT, 0,  ng,      lane, a00);
      store_c(T, LDT, 0,  ng + 16, lane, a01);
      store_c(T, LDT, 16, ng,      lane, a10);
      store_c(T, LDT, 16, ng + 16, lane, a11);
    }

    // x(b) fully consumed: prefetch next element's 16KB tile; the ~650-WMMA
    // Newton-Schulz phase below hides the DRAM latency (global_prefetch_b8).
    if (b + nWaves < nB) {
      const float* xn = x + (size_t)(b + nWaves) * 4096;
      for (int off = lane * 16; off < 4096; off += 32 * 16)
        __builtin_prefetch(xn + off, 0, 1);
    }

    // ---- Step 2: Y = T * W (symmetric: 3 tiles + mirror) ----
    {
      v8f y00 = {}, y01 = {}, y11 = {};
      for (int kb = 0; kb < 64; kb += 4) {
        v2f t0 = frag_row2(T, LDT, 0, kb, lane);      // ds_load_b64
        v2f t1 = frag_row2(T, LDT, 16, kb, lane);
        v2f w0 = frag_col(Wlds, LDW, kb, 0, lane);
        v2f w1 = frag_col(Wlds, LDW, kb, 16, lane);
        y00 = wmma4(t0, w0, y00);
        y01 = wmma4(t0, w1, y01);
        y11 = wmma4(t1, w1, y11);
      }
      store_c(Y, LD32, 0, 0, lane, y00);
      store_c(Y, LD32, 16, 16, lane, y11);
      store_c(Y, LD32, 0, 16, lane, y01);
      store_c_t(Y, LD32, 0, 16, lane, y01);
    }

    // ---- Step 3: Newton-Schulz sign(M), M = Y - EPS*I; Z0 = M/||M||_F ----
    float ss = 0.f;
    for (int i = lane; i < 1024; i += 32) {
      float v = Y[(i >> 5) * LD32 + (i & 31)] - ((i % 33) == 0 ? EPS : 0.f);
      ss += v * v;
    }
#pragma unroll
    for (int off = 16; off; off >>= 1) ss += __shfl_xor(ss, off, 32);
    float rn = ss > 0.f ? rsqrtf(ss) : 0.f;
    for (int i = lane; i < 1024; i += 32) {
      int idx = (i >> 5) * LD32 + (i & 31);
      P[idx] = (Y[idx] - ((i % 33) == 0 ? EPS : 0.f)) * rn;
    }

    for (int it = 0; it < NS_ITERS; ++it) {
      // ZZ = P*P : P symmetric, each b64 fragment is both A and B operand.
      v8f z00 = {}, z01 = {}, z11 = {};
      for (int kb = 0; kb < 32; kb += 4) {
        v2f p0 = frag_row2(P, LD32, 0, kb, lane);
        v2f p1 = frag_row2(P, LD32, 16, kb, lane);
        z00 = wmma4(p0, p0, z00);
        z01 = wmma4(p0, p1, z01);
        z11 = wmma4(p1, p1, z11);
      }
      store_c(ZZ, LD32, 0, 0, lane, z00);
      store_c(ZZ, LD32, 16, 16, lane, z11);
      store_c(ZZ, LD32, 0, 16, lane, z01);
      store_c_t(ZZ, LD32, 0, 16, lane, z01);

      // Q = 1.5*P - 0.5*P*ZZ : P, ZZ commute -> product symmetric (3 tiles).
      v8f q00 = {}, q01 = {}, q11 = {};
      for (int kb = 0; kb < 32; kb += 4) {
        v2f p0 = frag_row2(P, LD32, 0, kb, lane);
        v2f p1 = frag_row2(P, LD32, 16, kb, lane);
        v2f s0 = frag_row2(ZZ, LD32, 0, kb, lane);
        v2f s1 = frag_row2(ZZ, LD32, 16, kb, lane);
        q00 = wmma4(p0, s0, q00);
        q01 = wmma4(p0, s1, q01);
        q11 = wmma4(p1, s1, q11);
      }
      store_ns(Q, P, 0, 0, lane, q00, false);
      store_ns(Q, P, 16, 16, lane, q11, false);
      store_ns(Q, P, 0, 16, lane, q01, true);
      float* tsw = P; P = Q; Q = tsw;
    }

    // ---- Step 4: R = 0.5*(Y + EPS*I + S*M), S = P, M = Y - EPS*I (symmetric) ----
    {
      v8f r00 = {}, r01 = {}, r11 = {};
      for (int kb = 0; kb < 32; kb += 4) {
        v2f s0 = frag_row2(P, LD32, 0, kb, lane);
        v2f s1 = frag_row2(P, LD32, 16, kb, lane);
        v2f m0 = frag_row2(Y, LD32, 0, kb, lane);
        v2f m1 = frag_row2(Y, LD32, 16, kb, lane);
        int k  = kb + ((lane >> 4) << 1);
        int nA = (lane & 15);
        int nBv = 16 + nA;
        m0.x -= (k == nA)        ? EPS : 0.f;
        m0.y -= ((k + 1) == nA)  ? EPS : 0.f;
        m1.x -= (k == nBv)       ? EPS : 0.f;
        m1.y -= ((k + 1) == nBv) ? EPS : 0.f;
        r00 = wmma4(s0, m0, r00);
        r01 = wmma4(s0, m1, r01);
        r11 = wmma4(s1, m1, r11);
      }
      store_fin(Q, Y, 0, 0, lane, r00, false);
      store_fin(Q, Y, 16, 16, lane, r11, false);
      store_fin(Q, Y, 0, 16, lane, r01, true);
    }

    // ---- Step 5: classifier out[b] = flat(R) @ cls_w^T + cls_b ----
    float s0 = 0.f, s1 = 0.f;
    for (int i = lane; i < 1024; i += 32) {
      float f = Q[(i >> 5) * LD32 + (i & 31)];
      s0 += f * cls_w[i];
      s1 += f * cls_w[1024 + i];
    }
#pragma unroll
    for (int off = 16; off; off >>= 1) {
      s0 += __shfl_xor(s0, off, 32);
      s1 += __shfl_xor(s1, off, 32);
    }
    if (lane == 0) {
      out[2 * (size_t)b + 0] = s0 + cls_b[0];
      out[2 * (size_t)b + 1] = s1 + cls_b[1];
    }
  }
}

extern "C" void kernel_launch(void* const* d_in, const int* in_sizes, int n_in,
                              void* d_out, int out_size, void* d_ws, size_t ws_size,
                              hipStream_t stream) {
  const float* x     = (const float*)d_in[0];   // [B,64,64]
  const float* W     = (const float*)d_in[1];   // [64,32]
  const float* cls_w = (const float*)d_in[2];   // [2,1024]
  const float* cls_b = (const float*)d_in[3];   // [2]
  float* out = (float*)d_out;                   // [B,2]

  int nB = in_sizes[0] / (64 * 64);
  int wgs = NUM_WGS;
  int maxWgs = (nB + WAVES_PER_WG - 1) / WAVES_PER_WG;
  if (wgs > maxWgs) wgs = maxWgs;
  size_t shmem = (size_t)(64 * LDW + WAVES_PER_WG * (32 * LDT + 4 * 32 * LD32)) * sizeof(float);
  spdnet_wmma_kernel<<<wgs, WAVES_PER_WG * 32, shmem, stream>>>(x, W, cls_w, cls_b, out, nB);
}